// TemporalConsistencyLoss_13872744366785
// MI455X (gfx1250) — compile-verified
//
#include <hip/hip_runtime.h>

// Problem constants (match the reference).
#define TT 128   // time steps
#define NN 2048  // trajectories
#define DD 64    // latent dim
// loss scale: -1 / (N*T*T) = -1/33554432 (exact power of two in fp32)
#define NEG_INV_NTT (-1.0f / 33554432.0f)

typedef float v2f __attribute__((ext_vector_type(2)));
typedef float v8f __attribute__((ext_vector_type(8)));

__device__ __forceinline__ float waveReduceSum(float v) {
#pragma unroll
  for (int m = 16; m >= 1; m >>= 1) v += __shfl_xor(v, m, 32);
  return v;
}

// ---------------------------------------------------------------------------
// Stage 1: normalize rows and accumulate per-(array, chunk, n) partial sums.
// One wave per (arr, chunk, n). Lane l owns dims {2l, 2l+1} -> a wave reads
// each 256B row fully coalesced (32 x b64). Consecutive waves take
// consecutive n, so an 8-wave block streams 2KB contiguous per t step.
// ---------------------------------------------------------------------------
__global__ void __launch_bounds__(256)
tcl_stage1(const float* __restrict__ p, const float* __restrict__ z,
           float* __restrict__ part, int nchunk, int tchunk) {
  const int wave = blockIdx.x * (blockDim.x >> 5) + (threadIdx.x >> 5);
  const int lane = threadIdx.x & 31;
  const int n    = wave & (NN - 1);
  const int rest = wave >> 11;           // arr*nchunk + chunk
  const int chunk = rest & (nchunk - 1);
  const int arr   = rest / nchunk;       // 0 = p, 1 = z (wave-uniform)
  const float* __restrict__ x = arr ? z : p;

  float ax = 0.0f, ay = 0.0f;
  const int t0 = chunk * tchunk;
#pragma unroll 4
  for (int tt = 0; tt < tchunk; ++tt) {
    const int t = t0 + tt;
    const float2* __restrict__ row =
        (const float2*)(x + ((size_t)t * NN + n) * DD);
    float2 v = row[lane];
    float s = v.x * v.x + v.y * v.y;
    s = waveReduceSum(s);                       // ||row||^2 over all 64 dims
    float inv = 1.0f / fmaxf(sqrtf(s), 1e-12f); // F.normalize eps semantics
    ax += v.x * inv;
    ay += v.y * inv;
  }
  float2 acc; acc.x = ax; acc.y = ay;
  float2* __restrict__ dst =
      (float2*)(part + (((size_t)(arr * nchunk + chunk) * NN) + n) * DD);
  dst[lane] = acc;
}

// ---------------------------------------------------------------------------
// Stage 2: per 16-trajectory tile, D_tile = Sp(16x64) @ Sz^T(64x16) via
// 16 x V_WMMA_F32_16X16X4_F32; the diagonal D[m][m] = Sp[n0+m]·Sz[n0+m].
// A 16x4 f32 layout: lane L -> M = L&15, VGPR0 holds K = 2*(L>>4),
// VGPR1 holds K+1. B 4x16 mirrors it (row K striped across lanes), so both
// operands use the same gather pattern from their arrays.
// ---------------------------------------------------------------------------
__global__ void __launch_bounds__(128)
tcl_stage2(const float* __restrict__ part, float* __restrict__ tileSums,
           int nchunk) {
  const int wave = blockIdx.x * (blockDim.x >> 5) + (threadIdx.x >> 5);
  const int lane = threadIdx.x & 31;
  const int n0   = wave * 16;
  const int nIdx = n0 + (lane & 15);
  const int kHi  = (lane >> 4) * 2;  // 0 or 2

  const float* __restrict__ pPart = part;
  const float* __restrict__ zPart = part + (size_t)nchunk * NN * DD;

  v8f c = {0.f, 0.f, 0.f, 0.f, 0.f, 0.f, 0.f, 0.f};
#pragma unroll
  for (int kb = 0; kb < DD / 4; ++kb) {
    const int k0 = kb * 4 + kHi;
    float a0 = 0.f, a1 = 0.f, b0 = 0.f, b1 = 0.f;
    for (int ch = 0; ch < nchunk; ++ch) {   // chunk-reduce the partials inline
      const float* pb = pPart + ((size_t)ch * NN + nIdx) * DD + k0;
      const float* zb = zPart + ((size_t)ch * NN + nIdx) * DD + k0;
      a0 += pb[0]; a1 += pb[1];
      b0 += zb[0]; b1 += zb[1];
    }
    v2f a; a[0] = a0; a[1] = a1;
    v2f b; b[0] = b0; b[1] = b1;
    // (neg_a, A, neg_b, B, c_mod, C, reuse_a, reuse_b)
    c = __builtin_amdgcn_wmma_f32_16x16x4_f32(false, a, false, b,
                                              (short)0, c, false, false);
  }

  // Diagonal: m<8 -> lane m, vgpr m; m>=8 -> lane m+16 (24..31), vgpr m-8.
  float diag = 0.f;
#pragma unroll
  for (int r = 0; r < 8; ++r) {
    float cv = c[r];
    diag += ((lane == r) || (lane == 24 + r)) ? cv : 0.f;
  }
  float s = waveReduceSum(diag);
  if (lane == 0) tileSums[wave] = s;
}

// ---------------------------------------------------------------------------
// Stage 3: reduce the 128 tile sums, scale, write scalar loss.
// ---------------------------------------------------------------------------
__global__ void __launch_bounds__(128)
tcl_stage3(const float* __restrict__ tileSums, float* __restrict__ out,
           int ntiles) {
  __shared__ float sm[4];
  const int tid = threadIdx.x, lane = tid & 31, w = tid >> 5;
  float v = (tid < ntiles) ? tileSums[tid] : 0.f;
  v = waveReduceSum(v);
  if (lane == 0) sm[w] = v;
  __syncthreads();
  if (w == 0) {
    float x = (lane < 4) ? sm[lane] : 0.f;
    x = waveReduceSum(x);
    if (lane == 0) out[0] = x * NEG_INV_NTT;
  }
}

extern "C" void kernel_launch(void* const* d_in, const int* in_sizes, int n_in,
                              void* d_out, int out_size, void* d_ws,
                              size_t ws_size, hipStream_t stream) {
  (void)in_sizes; (void)n_in; (void)out_size;
  const float* p = (const float*)d_in[0];
  const float* z = (const float*)d_in[1];
  // d_in[2] = done: dead code in the reference's effective math.
  float* out = (float*)d_out;

  // Pick the number of time-chunks (more = more waves in flight for the
  // bandwidth stage) that fits the provided workspace.
  int nchunk = 8;
  const int ntiles = NN / 16;  // 128
  while (nchunk > 1 &&
         (size_t)2 * nchunk * NN * DD * sizeof(float) +
                 ntiles * sizeof(float) > ws_size)
    nchunk >>= 1;
  const int tchunk = TT / nchunk;

  float* part = (float*)d_ws;                       // [2][nchunk][N][D] fp32
  float* tileSums = part + (size_t)2 * nchunk * NN * DD;

  const int totalWaves = 2 * nchunk * NN;           // stage-1 waves
  tcl_stage1<<<dim3(totalWaves / 8), dim3(256), 0, stream>>>(p, z, part,
                                                             nchunk, tchunk);
  tcl_stage2<<<dim3(ntiles / 4), dim3(128), 0, stream>>>(part, tileSums,
                                                         nchunk);
  tcl_stage3<<<dim3(1), dim3(128), 0, stream>>>(tileSums, out, ntiles);
}